// ConvertTRTFormatToFlatTensor_83537113907302
// MI455X (gfx1250) — compile-verified
//
#include <hip/hip_runtime.h>
#include <hip/hip_bf16.h>

// ConvertTRTFormatToFlatTensor for MI455X (gfx1250).
//
// Structural insight: the mask (m < num_predictions[b]) is a contiguous prefix
// of each batch, so stable stream compaction == per-batch shifted copy:
//   dst(b, m) = excl_scan(num_predictions)[b] + m        for m < n_b
// Rows >= total are zero. Pure bandwidth problem (~100 MB @ 23.3 TB/s ~ 4.5us).
//
// CDNA5 paths used:
//  - global_load_async_to_lds_b128  (memory -> LDS async copy of box quads)
//  - global_store_async_from_lds_b32 (LDS -> memory async copy of output rows,
//    no VGPR round-trip; drained by S_ENDPGM implicit wait-idle)
//  - s_wait_asynccnt for the load->use dependency
//  - wave32-native 256-thread blocks, LDS workgroup scan (ds ops),
//    stride-7 LDS rows (7 coprime to 64 banks -> conflict-free)

#define TPB 256

__global__ void trt_scan_offsets_kernel(const int* __restrict__ nump,
                                        int* __restrict__ offs,      // [B+1]
                                        int* __restrict__ count_out, // raw int32 into d_out tail
                                        int M) {
  __shared__ int s[TPB];
  const int B = blockDim.x;       // launched with B threads (B == 256)
  const int t = threadIdx.x;

  int v = nump[t];
  v = v < 0 ? 0 : (v > M ? M : v);   // defensive clamp, matches mask semantics
  s[t] = v;
  __syncthreads();

  // Hillis-Steele inclusive scan over B elements
  for (int off = 1; off < B; off <<= 1) {
    int x = (t >= off) ? s[t - off] : 0;
    __syncthreads();
    s[t] += x;
    __syncthreads();
  }

  const int incl = s[t];
  offs[t] = incl - v;               // exclusive prefix
  if (t == B - 1) {
    offs[B] = incl;                 // total valid count N
    count_out[0] = incl;            // second reference output (int32 bits)
  }
}

__global__ void trt_scatter_compact_kernel(const int* __restrict__ nump,
                                           const float4* __restrict__ boxes,  // [B*M] float4
                                           const float* __restrict__ scores,  // [B*M]
                                           const int* __restrict__ classes,   // [B*M]
                                           const int* __restrict__ offs,      // [B+1]
                                           float* __restrict__ out,           // [B*M*7]
                                           int M, int blocksPerBatch, int B) {
  __shared__ __align__(16) float4 sbox[TPB];   // async-load staging
  __shared__ __align__(16) float srow[TPB * 7]; // row transpose buffer

  const int tid = threadIdx.x;
  const int b   = blockIdx.x / blocksPerBatch;        // uniform per block
  const int m0  = (blockIdx.x % blocksPerBatch) * TPB;
  const int m   = m0 + tid;
  const int r0  = b * M + m0;                          // <= 2^21, fits int
  const int r   = r0 + tid;

  int n = nump[b];
  n = n < 0 ? 0 : (n > M ? M : n);
  const int total = offs[B];
  const bool valid = (m < n);

  // Kick off async 16B box load into this lane's LDS slot (ASYNCcnt path).
  if (valid) {
    const unsigned lds_off = (unsigned)(uintptr_t)(&sbox[tid]); // low 32 bits == LDS offset
    const unsigned goff    = (unsigned)r * 16u;                 // byte offset into boxes
    asm volatile("global_load_async_to_lds_b128 %0, %1, %2"
                 :
                 : "v"(lds_off), "v"(goff),
                   "s"((unsigned long long)(uintptr_t)boxes)
                 : "memory");
  }

  // Overlap with in-flight async loads: zero-fill dest rows in
  // [max(total, r0), r0+TPB). Contiguous span -> unit-stride coalesced stores.
  {
    int z0 = total - r0;            // first local row to zero
    if (z0 < TPB) {
      if (z0 < 0) z0 = 0;
      const size_t base = ((size_t)(r0 + z0)) * 7u;
      const int nfl = (TPB - z0) * 7;
      for (int i = tid; i < nfl; i += TPB) out[base + i] = 0.0f;
    }
  }

  // Overlap: pull the scalar fields while the async load is in flight.
  float sc = 0.0f, cl = 0.0f;
  if (valid) {
    sc = scores[r];
    cl = (float)classes[r];
  }

  // Wait for this wave's async LDS writes (each lane reads only its own slot).
  asm volatile("s_wait_asynccnt 0" ::: "memory");

  if (valid) {
    const float4 bx = sbox[tid];
    float* sr = &srow[tid * 7];
    sr[0] = (float)b;   // image_index
    sr[1] = bx.x;
    sr[2] = bx.y;
    sr[3] = bx.z;
    sr[4] = bx.w;
    sr[5] = sc;         // score
    sr[6] = cl;         // class
  }
  __syncthreads();   // all ds_stores to srow complete before any async store reads it

  // Valid rows of this block map to the contiguous dest span
  // [offs[b]+m0, offs[b]+m0+nvalid). Stream LDS -> global directly with
  // async stores (no VGPR round-trip); S_ENDPGM's implicit wait-idle drains
  // ASYNCcnt, and srow is not touched again, so no explicit wait is needed.
  int nvalid = n - m0;
  if (nvalid > 0) {
    if (nvalid > TPB) nvalid = TPB;
    const unsigned srow_base = (unsigned)(uintptr_t)(&srow[0]);
    const unsigned gbase     = (unsigned)(((size_t)(offs[b] + m0)) * 28u); // byte offset
    const int nfl = nvalid * 7;
    for (int i = tid; i < nfl; i += TPB) {
      const unsigned lds_a = srow_base + (unsigned)i * 4u;
      const unsigned gof   = gbase + (unsigned)i * 4u;
      asm volatile("global_store_async_from_lds_b32 %0, %1, %2"
                   :
                   : "v"(gof), "v"(lds_a),
                     "s"((unsigned long long)(uintptr_t)out)
                   : "memory");
    }
  }
}

extern "C" void kernel_launch(void* const* d_in, const int* in_sizes, int n_in,
                              void* d_out, int out_size, void* d_ws, size_t ws_size,
                              hipStream_t stream) {
  const int*    nump    = (const int*)d_in[0];     // [B,1] int32
  const float4* boxes   = (const float4*)d_in[1];  // [B,M,4] f32 -> float4 rows
  const float*  scores  = (const float*)d_in[2];   // [B,M] f32
  const int*    classes = (const int*)d_in[3];     // [B,M] int32
  float*        out     = (float*)d_out;           // [B*M*7] f32 + 1 count slot

  const int B = in_sizes[0];          // 256
  const int M = in_sizes[2] / B;      // 8192 (multiple of TPB)

  int* offs = (int*)d_ws;                                        // B+1 ints of scratch
  int* count_out = (int*)(out + (size_t)B * (size_t)M * 7u);     // raw int32 bits for N

  trt_scan_offsets_kernel<<<1, B, 0, stream>>>(nump, offs, count_out, M);

  const int blocksPerBatch = M / TPB;
  trt_scatter_compact_kernel<<<B * blocksPerBatch, TPB, 0, stream>>>(
      nump, boxes, scores, classes, offs, out, M, blocksPerBatch, B);
}